// NonLinearReadoutLayer_29850022707548
// MI455X (gfx1250) — compile-verified
//
#include <hip/hip_runtime.h>

typedef __attribute__((ext_vector_type(2))) float v2f;
typedef __attribute__((ext_vector_type(8))) float v8f;

#define XDIM 512      // 128 scalar + 384 vector features per row
#define OUTD 64       // 16 + 48
#define NROWS 200000
#define NTILES (NROWS / 16)

// ---- LDS layout (dwords). Weights stored PAIR-INTERLEAVED for WMMA B-frags:
//   lds[pair p][2n+0] = w[2p+0][n],  lds[pair p][2n+1] = w[2p+1][n]
// so a B fragment (rows 4k+2h, 4k+2h+1 ; col n) is ONE aligned ds_load_b64.
// Pair-row strides ≡ 32 (mod 64 banks): the two lane-halves (p vs p+1) hit
// disjoint bank halves -> conflict-free b64 loads.
#define P1S_STRIDE 544   // 2*256 + 32
#define P1V_STRIDE 288   // 2*128 + 32
#define P2_STRIDE  32    // 2*16  (32 % 64 == 32 already)
#define SCR_STRIDE 132   // 128 + 4 (row-major act/gated scratch, b64-friendly)
#define OFF_W1S 0
#define OFF_W1V (OFF_W1S + 64 * P1S_STRIDE)                  // 34816
#define OFF_W2S (OFF_W1V + 64 * P1V_STRIDE)                  // 53248
#define OFF_W2V (OFF_W2S + 64 * P2_STRIDE)                   // 55296
#define OFF_SCR (OFF_W2V + 64 * P2_STRIDE)                   // 57344
#define LDS_DW  (OFF_SCR + 8 * 16 * SCR_STRIDE)              // 74240 dw = 290 KB

__device__ __forceinline__ v8f wmma_f32(v2f a, v2f b, v8f c) {
  return __builtin_amdgcn_wmma_f32_16x16x4_f32(
      /*neg_a=*/false, a, /*neg_b=*/false, b,
      /*c_mod=*/(short)0, c, /*reuse_a=*/false, /*reuse_b=*/false);
}

__device__ __forceinline__ v2f lds_b64(const float* p) {
  return *(const v2f*)p;
}

__global__ __launch_bounds__(256) void
nlr_kernel(const float* __restrict__ x,
           const float* __restrict__ w1_s,
           const float* __restrict__ w1_v,
           const float* __restrict__ w2_s,
           const float* __restrict__ w2_v,
           float* __restrict__ out)
{
  extern __shared__ float lds[];
  const int tid = threadIdx.x;

  // ---- cooperative weight staging: row-major global -> pair-interleaved LDS ----
  for (int i = tid; i < 128 * 256; i += 256) {
    int r = i >> 8, c = i & 255;
    lds[OFF_W1S + (r >> 1) * P1S_STRIDE + 2 * c + (r & 1)] = w1_s[i];
  }
  for (int i = tid; i < 128 * 128; i += 256) {
    int r = i >> 7, c = i & 127;
    lds[OFF_W1V + (r >> 1) * P1V_STRIDE + 2 * c + (r & 1)] = w1_v[i];
  }
  for (int i = tid; i < 128 * 16; i += 256) {
    int r = i >> 4, c = i & 15;
    lds[OFF_W2S + (r >> 1) * P2_STRIDE + 2 * c + (r & 1)] = w2_s[i];
    lds[OFF_W2V + (r >> 1) * P2_STRIDE + 2 * c + (r & 1)] = w2_v[i];
  }
  __syncthreads();

  const int wave = tid >> 5;       // 8 waves per WG (wave32)
  const int lane = tid & 31;
  const int half = lane >> 4;      // K-half selector for A/B fragments
  const int lid  = lane & 15;      // M (A) / N (B,C) index within fragment
  float* scr = &lds[OFF_SCR + wave * 16 * SCR_STRIDE];   // per-wave 16x132 scratch

  const float inv = 0.08838834764831845f;   // 1/sqrt(128)

  const int wavesTotal = (int)gridDim.x * 8;
  for (int tile = (int)blockIdx.x * 8 + wave; tile < NTILES; tile += wavesTotal) {
    const int row0 = tile * 16;

    // ================= GEMM 1a: h_s = s @ w1_s  (16x256, K=128) =================
    const float* xs = x + (size_t)(row0 + lid) * XDIM + 2 * half;
    v8f accS[16];
#pragma unroll
    for (int t = 0; t < 16; t++) accS[t] = (v8f)0.0f;

#pragma unroll 4
    for (int k = 0; k < 32; k++) {
      v2f a = *(const v2f*)(xs + 4 * k);                       // A: row lid, K-pair
      const float* bp = &lds[OFF_W1S + (2 * k + half) * P1S_STRIDE + 2 * lid];
#pragma unroll
      for (int t = 0; t < 16; t++) {
        v2f b = lds_b64(bp + 32 * t);                          // one ds_load_b64
        accS[t] = wmma_f32(a, b, accS[t]);
      }
    }

    // ---- activations: act = silu(h_s[:, :128]); gates = sigmoid(h_s[:, 128:]) ----
    v8f gate[8];
#pragma unroll
    for (int t = 0; t < 8; t++) {
#pragma unroll
      for (int r = 0; r < 8; r++) {
        float hs  = accS[t][r] * inv;
        float act = hs / (1.0f + __expf(-hs));                 // silu
        float hg  = accS[8 + t][r] * inv;
        gate[t][r] = 1.0f / (1.0f + __expf(-hg));              // sigmoid
        scr[(r + 8 * half) * SCR_STRIDE + 16 * t + lid] = act; // C-layout -> row-major
      }
    }
    asm volatile("s_wait_dscnt 0" ::: "memory");               // scratch visible to wave

    // ================= GEMM 2a: out_s = act @ w2_s  (16x16, K=128) ==============
    v8f accO = (v8f)0.0f;
#pragma unroll 4
    for (int k = 0; k < 32; k++) {
      v2f a = lds_b64(scr + lid * SCR_STRIDE + 4 * k + 2 * half);
      v2f b = lds_b64(&lds[OFF_W2S + (2 * k + half) * P2_STRIDE + 2 * lid]);
      accO = wmma_f32(a, b, accO);
    }
    {
      float* od = out + (size_t)(row0 + 8 * half) * OUTD + lid;
#pragma unroll
      for (int r = 0; r < 8; r++) od[(size_t)r * OUTD] = accO[r] * inv;
    }

    // ====== GEMM 1b (fused over channels): h_v_c = v[:,:,c] @ w1_v, c=0..2 ======
    // Lane needs dwords 12k+6h+{0..5}: a contiguous run -> 3 x global_load_b64,
    // and every B fragment is reused by 3 WMMAs (one per channel).
    v8f accV[3][8];
#pragma unroll
    for (int c = 0; c < 3; c++)
#pragma unroll
      for (int t = 0; t < 8; t++) accV[c][t] = (v8f)0.0f;

    const float* xv = x + (size_t)(row0 + lid) * XDIM + 128 + 6 * half;
#pragma unroll 2
    for (int k = 0; k < 32; k++) {
      v2f q0 = *(const v2f*)(xv + 12 * k);        // dwords 0,1
      v2f q1 = *(const v2f*)(xv + 12 * k + 2);    // dwords 2,3
      v2f q2 = *(const v2f*)(xv + 12 * k + 4);    // dwords 4,5
      v2f a0; a0.x = q0.x; a0.y = q1.y;           // c=0: m-pair (d0, d3)
      v2f a1; a1.x = q0.y; a1.y = q2.x;           // c=1: (d1, d4)
      v2f a2; a2.x = q1.x; a2.y = q2.y;           // c=2: (d2, d5)
      const float* bp = &lds[OFF_W1V + (2 * k + half) * P1V_STRIDE + 2 * lid];
#pragma unroll
      for (int t = 0; t < 8; t++) {
        v2f b = lds_b64(bp + 32 * t);
        accV[0][t] = wmma_f32(a0, b, accV[0][t]);
        accV[1][t] = wmma_f32(a1, b, accV[1][t]);
        accV[2][t] = wmma_f32(a2, b, accV[2][t]);
      }
    }

    // ====== per channel: gate -> scratch -> GEMM 2b: out_v_c = gated @ w2_v =====
#pragma unroll
    for (int c = 0; c < 3; c++) {
      asm volatile("" ::: "memory");   // WAR on scratch; same-wave LDS is in-order
#pragma unroll
      for (int t = 0; t < 8; t++)
#pragma unroll
        for (int r = 0; r < 8; r++)
          scr[(r + 8 * half) * SCR_STRIDE + 16 * t + lid] = accV[c][t][r] * inv * gate[t][r];
      asm volatile("s_wait_dscnt 0" ::: "memory");

      v8f accW = (v8f)0.0f;
#pragma unroll 4
      for (int k = 0; k < 32; k++) {
        v2f a = lds_b64(scr + lid * SCR_STRIDE + 4 * k + 2 * half);
        v2f b = lds_b64(&lds[OFF_W2V + (2 * k + half) * P2_STRIDE + 2 * lid]);
        accW = wmma_f32(a, b, accW);
      }
      float* ov = out + (size_t)(row0 + 8 * half) * OUTD + 16 + 3 * lid + c;
#pragma unroll
      for (int r = 0; r < 8; r++) ov[(size_t)r * OUTD] = accW[r] * inv;
    }
  }
}

extern "C" void kernel_launch(void* const* d_in, const int* in_sizes, int n_in,
                              void* d_out, int out_size, void* d_ws, size_t ws_size,
                              hipStream_t stream) {
  const float* x    = (const float*)d_in[0];
  const float* w1_s = (const float*)d_in[1];
  const float* w1_v = (const float*)d_in[2];
  const float* w2_s = (const float*)d_in[3];
  const float* w2_v = (const float*)d_in[4];
  float* out = (float*)d_out;
  (void)in_sizes; (void)n_in; (void)out_size; (void)d_ws; (void)ws_size;

  const size_t ldsBytes = (size_t)LDS_DW * sizeof(float);     // ~290 KB (<= 320 KB/WGP)
  (void)hipFuncSetAttribute((const void*)nlr_kernel,
                            hipFuncAttributeMaxDynamicSharedMemorySize,
                            (int)ldsBytes);
  dim3 grid(512), block(256);
  nlr_kernel<<<grid, block, ldsBytes, stream>>>(x, w1_s, w1_v, w2_s, w2_v, out);
}